// GraphConvolutionII_35321811042822
// MI455X (gfx1250) — compile-verified
//
#include <hip/hip_runtime.h>

#define ALPHA 0.9f

typedef __attribute__((ext_vector_type(2))) float v2f;
typedef __attribute__((ext_vector_type(8))) float v8f;

// ---------------------------------------------------------------------------
// Kernel 1: h = (1-ALPHA) * (x_initial @ w_init)   [n, 256]
// Writes every element of the workspace -> also the zero/init pass for the
// edge scatter.
// ---------------------------------------------------------------------------
__global__ void gcn2_init_kernel(const float* __restrict__ xi,   // [n,3]
                                 const float* __restrict__ wi,   // [3,256]
                                 float* __restrict__ h,          // [n,256]
                                 int n) {
  const int total4 = n * 64;                       // float4 elements
  const float4* w4 = (const float4*)wi;            // [3,64] float4
  float4* h4 = (float4*)h;
  const float beta = 1.0f - ALPHA;
  for (int idx = blockIdx.x * blockDim.x + threadIdx.x; idx < total4;
       idx += gridDim.x * blockDim.x) {
    const int node = idx >> 6;
    const int f4 = idx & 63;
    const float x0 = xi[node * 3 + 0];
    const float x1 = xi[node * 3 + 1];
    const float x2 = xi[node * 3 + 2];
    const float4 w0 = w4[f4];
    const float4 w1 = w4[64 + f4];
    const float4 w2 = w4[128 + f4];
    float4 r;
    r.x = beta * (x0 * w0.x + x1 * w1.x + x2 * w2.x);
    r.y = beta * (x0 * w0.y + x1 * w1.y + x2 * w2.y);
    r.z = beta * (x0 * w0.z + x1 * w1.z + x2 * w2.z);
    r.w = beta * (x0 * w0.w + x1 * w1.w + x2 * w2.w);
    h4[idx] = r;
  }
}

// ---------------------------------------------------------------------------
// Kernel 2: COO SpMM scatter: h[row] += ALPHA * val * x[col]
// One wave32 per edge; lane l owns 8 contiguous floats of the 256-wide row.
// x (102 MB) is L2-resident on MI455X (192 MB L2) -> gathers hit L2.
// unsafeAtomicAdd lowers to no-return global_atomic_add_f32 (STOREcnt).
// ---------------------------------------------------------------------------
__global__ void gcn2_scatter_kernel(const float* __restrict__ x,     // [n,256]
                                    const int* __restrict__ erow,
                                    const int* __restrict__ ecol,
                                    const float* __restrict__ eval_,
                                    float* __restrict__ h,           // [n,256]
                                    int nedges) {
  const int lane = threadIdx.x & 31;
  const int gwave = (blockIdx.x * blockDim.x + threadIdx.x) >> 5;
  const int nwaves = (gridDim.x * blockDim.x) >> 5;
  for (int e = gwave; e < nedges; e += nwaves) {
    const int r = erow[e];
    const int c = ecol[e];
    const float a = ALPHA * eval_[e];
    const float4* xs = (const float4*)(x + (size_t)c * 256) + (lane << 1);
    const float4 v0 = xs[0];
    const float4 v1 = xs[1];
    float* dst = h + (size_t)r * 256 + (lane << 3);
    unsafeAtomicAdd(dst + 0, a * v0.x);
    unsafeAtomicAdd(dst + 1, a * v0.y);
    unsafeAtomicAdd(dst + 2, a * v0.z);
    unsafeAtomicAdd(dst + 3, a * v0.w);
    unsafeAtomicAdd(dst + 4, a * v1.x);
    unsafeAtomicAdd(dst + 5, a * v1.y);
    unsafeAtomicAdd(dst + 6, a * v1.z);
    unsafeAtomicAdd(dst + 7, a * v1.w);
  }
}

// ---------------------------------------------------------------------------
// Kernel 3: out = relu(h @ w_x) using V_WMMA_F32_16X16X4_F32 (exact fp32).
// One wave per block; each wave owns a 32x256 slab of h (two 16-row A tiles)
// staged in LDS with row stride 260 floats (conflict-free ds_load_b64 reads).
// Both A tiles share every B fragment -> 1 B load per WMMA, halved w_x L2
// traffic vs one-tile-per-wave.
// A layout (ISA 16x4 f32): lanes 0-15 -> K={0,1}, lanes 16-31 -> K={2,3}.
// B layout (4x16): VGPR0 lanes 0-15 K=0 / lanes 16-31 K=2; VGPR1 K=1 / K=3.
// C/D layout: VGPR r -> M=r (lanes 0-15) and M=r+8 (lanes 16-31), N=lane%16.
// ---------------------------------------------------------------------------
__global__ __launch_bounds__(32) void gcn2_gemm_kernel(
    const float* __restrict__ h,    // [n,256]
    const float* __restrict__ wx,   // [256,256]
    float* __restrict__ out,        // [n,256]
    int n, int ntiles) {
  __shared__ float lh[32 * 260];    // 33,280 B
  const int lane = threadIdx.x & 31;
  const int tile0 = blockIdx.x * 2;
  if (tile0 >= ntiles) return;
  const int rowbase0 = tile0 * 16;
  const int rowbase1 = rowbase0 + 16;

  // Stage the 32x256 h slab into LDS (padded stride 260).
  for (int r = 0; r < 32; ++r) {
    int rr = rowbase0 + r;
    if (rr >= n) rr = n - 1;
    const float4* src = (const float4*)(h + (size_t)rr * 256) + (lane << 1);
    const float4 a0 = src[0];
    const float4 a1 = src[1];
    *(float4*)&lh[r * 260 + (lane << 3)] = a0;
    *(float4*)&lh[r * 260 + (lane << 3) + 4] = a1;
  }

  const int lrow = lane & 15;           // M (for A/D) or N (for B/D) index
  const int khalf = (lane >> 4) << 1;   // 0 or 2: K sub-offset for this half
  const int mhalf = (lane >> 4) << 3;   // 0 or 8: M offset of D rows
  const float* ap0base = &lh[lrow * 260];
  const float* ap1base = ap0base + 16 * 260;

  const bool full0 = (rowbase0 + 16 <= n);              // wave-uniform
  const bool full1 = (tile0 + 1 < ntiles) && (rowbase1 + 16 <= n);

  for (int j = 0; j < 16; ++j) {        // 16 output column tiles of width 16
    v8f acc0 = {};
    v8f acc1 = {};
    const float* bcol = wx + (j << 4) + lrow;
#pragma unroll 4
    for (int k4 = 0; k4 < 64; ++k4) {   // K = 256 in steps of 4
      const int kk = (k4 << 2) + khalf;
      v2f a0, a1, b;
      a0[0] = ap0base[kk];
      a0[1] = ap0base[kk + 1];
      a1[0] = ap1base[kk];
      a1[1] = ap1base[kk + 1];
      const float* bp = bcol + (size_t)kk * 256;
      b[0] = bp[0];
      b[1] = bp[256];
      acc0 = __builtin_amdgcn_wmma_f32_16x16x4_f32(false, a0, false, b,
                                                   (short)0, acc0, false, false);
      acc1 = __builtin_amdgcn_wmma_f32_16x16x4_f32(false, a1, false, b,
                                                   (short)0, acc1, false, false);
    }
    const int ncol = (j << 4) + lrow;
    // Tile 0 stores (hot path is branch-free: rowbase checks are uniform).
    if (full0) {
#pragma unroll
      for (int r = 0; r < 8; ++r)
        out[(size_t)(rowbase0 + mhalf + r) * 256 + ncol] = fmaxf(acc0[r], 0.0f);
    } else {
#pragma unroll
      for (int r = 0; r < 8; ++r) {
        const int m = rowbase0 + mhalf + r;
        if (m < n) out[(size_t)m * 256 + ncol] = fmaxf(acc0[r], 0.0f);
      }
    }
    // Tile 1 stores.
    if (full1) {
#pragma unroll
      for (int r = 0; r < 8; ++r)
        out[(size_t)(rowbase1 + mhalf + r) * 256 + ncol] = fmaxf(acc1[r], 0.0f);
    } else if (tile0 + 1 < ntiles) {
#pragma unroll
      for (int r = 0; r < 8; ++r) {
        const int m = rowbase1 + mhalf + r;
        if (m < n) out[(size_t)m * 256 + ncol] = fmaxf(acc1[r], 0.0f);
      }
    }
  }
}

// ---------------------------------------------------------------------------
extern "C" void kernel_launch(void* const* d_in, const int* in_sizes, int n_in,
                              void* d_out, int out_size, void* d_ws,
                              size_t ws_size, hipStream_t stream) {
  const float* x    = (const float*)d_in[0];   // [n, 256]
  const float* xi   = (const float*)d_in[1];   // [n, 3]
  const int*   erow = (const int*)d_in[2];     // [E]
  const int*   ecol = (const int*)d_in[3];     // [E]
  const float* ev   = (const float*)d_in[4];   // [E]
  const float* wi   = (const float*)d_in[5];   // [3, 256]
  const float* wx   = (const float*)d_in[6];   // [256, 256]
  float* out = (float*)d_out;

  const int n = in_sizes[0] / 256;
  const int nedges = in_sizes[2];
  float* hbuf = (float*)d_ws;                  // n*256 floats of scratch

  // 1) h = (1-alpha) * x_init @ w_init   (writes every element of hbuf)
  {
    const int total4 = n * 64;
    const int blocks = (total4 + 255) / 256;
    gcn2_init_kernel<<<blocks, 256, 0, stream>>>(xi, wi, hbuf, n);
  }
  // 2) h += alpha * (adj @ x)  (one wave per edge, 8 waves per 256-thr block)
  {
    const int blocks = (nedges + 7) / 8;
    gcn2_scatter_kernel<<<blocks, 256, 0, stream>>>(x, erow, ecol, ev, hbuf,
                                                    nedges);
  }
  // 3) out = relu(h @ w_x) via fp32 WMMA, 32 rows per wave/block
  {
    const int ntiles = (n + 15) / 16;
    const int blocks = (ntiles + 1) / 2;
    gcn2_gemm_kernel<<<blocks, 32, 0, stream>>>(hbuf, wx, out, n, ntiles);
  }
}